// CustomLocalNormalizedCrossCorrelationLoss_73641509257787
// MI455X (gfx1250) — compile-verified
//
#include <hip/hip_runtime.h>

typedef float v2f __attribute__((ext_vector_type(2)));
typedef float v8f __attribute__((ext_vector_type(8)));

#define HW          512
#define WIN         9
#define PADW        4          // WIN/2
#define TILE        16
#define HALO        24         // TILE + WIN - 1
#define STRIDE_X    25         // HALO + 1 (bank-conflict padding, 64 banks)
#define STRIDE_T    17         // TILE + 1
#define WAVES_PB    8
#define PER_WAVE    (2 * HALO * STRIDE_X + HALO * STRIDE_T)   // 600+600+408 = 1608 floats
#define N_TILES     (32 * 32 * 32)                            // 32768
#define N_BLOCKS    (N_TILES / WAVES_PB)                      // 4096
#define N_PIXELS    8388608.0f                                // 32*512*512

// D(16x16,f32) = A(16x4,f32) * B(4x16,f32) + C
#define WMMA_F32(a, b, c) \
    __builtin_amdgcn_wmma_f32_16x16x4_f32(false, (a), false, (b), (short)0, (c), false, false)

__global__ __launch_bounds__(256) void lncc_tile_kernel(const float* __restrict__ tpl,
                                                        const float* __restrict__ img,
                                                        float* __restrict__ partial) {
  __shared__ float lds[WAVES_PB * PER_WAVE];
  __shared__ float wavesum[WAVES_PB];

  const int lane   = threadIdx.x & 31;
  const int wave   = threadIdx.x >> 5;
  const int laneLo = lane & 15;
  const int laneHi = lane >> 4;

  // tile decode: image / tile-row / tile-col
  const int tileId = blockIdx.x * WAVES_PB + wave;
  const int imgIdx = tileId >> 10;          // / (32*32 tiles per image)
  const int rem    = tileId & 1023;
  const int ty     = rem >> 5;
  const int tx     = rem & 31;
  const int row0   = ty * TILE - PADW;      // halo origin (may be negative)
  const int col0   = tx * TILE - PADW;

  float* tplS = lds + wave * PER_WAVE;
  float* imgS = tplS + HALO * STRIDE_X;
  float* thS  = imgS + HALO * STRIDE_X;

  const long long base = (long long)imgIdx * (HW * HW);

  // ---- load 24x24 halo'd tile of both inputs, zero outside image ----
  #pragma unroll
  for (int it = 0; it < 18; ++it) {          // 576 elements / 32 lanes
    const int idx = it * 32 + lane;
    const int r = idx / HALO;
    const int c = idx - r * HALO;
    const int gr = row0 + r, gc = col0 + c;
    float tv = 0.0f, iv = 0.0f;
    if (gr >= 0 && gr < HW && gc >= 0 && gc < HW) {
      const long long off = base + (long long)gr * HW + gc;
      tv = tpl[off];
      iv = img[off];
    }
    tplS[r * STRIDE_X + c] = tv;
    imgS[r * STRIDE_X + c] = iv;
  }
  // wave-private LDS region; same-wave DS ops are in-order (ISA 7.3) -> no barrier

  v8f S[5];
  #pragma unroll
  for (int map = 0; map < 5; ++map) {
    // ---- stage 1: Th(24x16) = X(24x24) * W(24x16), W[k][j] = (j<=k<=j+8) ----
    #pragma unroll
    for (int mc = 0; mc < 2; ++mc) {         // row chunks [0,16) and [8,24)
      const int mb = mc * 8;
      v8f acc = {0.f, 0.f, 0.f, 0.f, 0.f, 0.f, 0.f, 0.f};
      for (int kc = 0; kc < 6; ++kc) {       // K = 24 in chunks of 4
        const int k0 = kc * 4;
        v2f a, b;
        #pragma unroll
        for (int v = 0; v < 2; ++v) {
          const int k  = k0 + v + 2 * laneHi;     // A/B layout: K = v + 2*(lane>>4)
          const int xr = mb + laneLo;             // A layout:   M = lane&15
          const float t = tplS[xr * STRIDE_X + k];
          const float m = imgS[xr * STRIDE_X + k];
          const float x = (map == 0) ? t
                        : (map == 1) ? m
                        : (map == 2) ? t * t
                        : (map == 3) ? m * m
                        :              t * m;
          a[v] = x;
          b[v] = (laneLo <= k && k <= laneLo + 8) ? 1.0f : 0.0f;   // W band
        }
        acc = WMMA_F32(a, b, acc);
      }
      // D layout: element (M = v + 8*laneHi + mb, N = laneLo); chunk 1 stores rows 16..23 only
      #pragma unroll
      for (int v = 0; v < 8; ++v) {
        const int r = mb + v + 8 * laneHi;
        if (mc == 0 || laneHi == 1) thS[r * STRIDE_T + laneLo] = acc[v];
      }
    }

    // ---- stage 2: S(16x16) = V(16x24) * Th(24x16), V[i][r] = (i<=r<=i+8) ----
    v8f acc2 = {0.f, 0.f, 0.f, 0.f, 0.f, 0.f, 0.f, 0.f};
    for (int kc = 0; kc < 6; ++kc) {
      const int k0 = kc * 4;
      v2f a, b;
      #pragma unroll
      for (int v = 0; v < 2; ++v) {
        const int k = k0 + v + 2 * laneHi;
        a[v] = (laneLo <= k && k <= laneLo + 8) ? 1.0f : 0.0f;  // V band, M = laneLo
        b[v] = thS[k * STRIDE_T + laneLo];                      // B layout
      }
      acc2 = WMMA_F32(a, b, acc2);
    }
    S[map] = acc2;
  }

  // ---- combine per-pixel score (matches reference algebra) + reduce ----
  float local = 0.0f;
  #pragma unroll
  for (int v = 0; v < 8; ++v) {
    const float area = 81.0f;
    const float Is  = S[0][v], Js  = S[1][v];
    const float I2s = S[2][v], J2s = S[3][v], IJs = S[4][v];
    const float uI = Is / area, uJ = Js / area;
    const float cross = IJs - uJ * Is - uI * Js + uI * uJ * area;
    const float Iv = I2s - 2.0f * uI * Is + uI * uI * area;
    const float Jv = J2s - 2.0f * uJ * Js + uJ * uJ * area;
    local += cross / (__builtin_sqrtf(Iv) * __builtin_sqrtf(Jv) + 1e-6f);
  }
  #pragma unroll
  for (int off = 16; off > 0; off >>= 1) local += __shfl_xor(local, off, 32);

  if (lane == 0) wavesum[wave] = local;
  __syncthreads();
  if (threadIdx.x == 0) {
    float s = 0.0f;
    #pragma unroll
    for (int w = 0; w < WAVES_PB; ++w) s += wavesum[w];
    partial[blockIdx.x] = s;   // deterministic per-block partial (no float atomics)
  }
}

__global__ __launch_bounds__(256) void lncc_reduce_kernel(const float* __restrict__ partial,
                                                          float* __restrict__ out,
                                                          int nPartial, float invN) {
  __shared__ float sm[256];
  float acc = 0.0f;
  for (int i = threadIdx.x; i < nPartial; i += 256) acc += partial[i];  // fixed order
  sm[threadIdx.x] = acc;
  __syncthreads();
  for (int s = 128; s > 0; s >>= 1) {
    if (threadIdx.x < s) sm[threadIdx.x] += sm[threadIdx.x + s];
    __syncthreads();
  }
  if (threadIdx.x == 0) out[0] = 1.0f - sm[0] * invN;
}

extern "C" void kernel_launch(void* const* d_in, const int* in_sizes, int n_in,
                              void* d_out, int out_size, void* d_ws, size_t ws_size,
                              hipStream_t stream) {
  (void)in_sizes; (void)n_in; (void)out_size; (void)ws_size;
  const float* tpl = (const float*)d_in[0];   // "template"
  const float* img = (const float*)d_in[1];   // "image"
  float* partial = (float*)d_ws;              // 4096 floats = 16 KB scratch

  lncc_tile_kernel<<<N_BLOCKS, 256, 0, stream>>>(tpl, img, partial);
  lncc_reduce_kernel<<<1, 256, 0, stream>>>(partial, (float*)d_out, N_BLOCKS, 1.0f / N_PIXELS);
}